// GPT2PagedAttention_575525617892
// MI455X (gfx1250) — compile-verified
//
#include <hip/hip_runtime.h>
#include <math.h>

// GPT-2 paged attention decode step for MI455X (gfx1250).
// fp32 WMMA (v_wmma_f32_16x16x4_f32) + Tensor Data Mover (tensor_load_to_lds)
// double-buffered KV streaming, flash-decoding split across 256 waves.
//
// Roofline: KV read = 128 MB -> ~5.5us @ 23.3 TB/s; AI ~ 8 FLOP/B, so fp32
// WMMA keeps the kernel bandwidth-bound. TDM moves K/V tiles directly into
// LDS (no VGPR round trip), overlapped with compute via TENSORcnt.

typedef float v2f __attribute__((ext_vector_type(2)));
typedef float v8f __attribute__((ext_vector_type(8)));
typedef unsigned int u32x4 __attribute__((ext_vector_type(4)));
typedef int i32x4 __attribute__((ext_vector_type(4)));
typedef int i32x8 __attribute__((ext_vector_type(8)));

#define LQ 16
#define DMODEL 1024
#define NHEAD 16
#define HDIM 64
#define BLKSZ 16
#define T_TOT 16384
#define NSPLIT 16
#define CHUNK (T_TOT / NSPLIT) /* 1024 */
#define ATT_SCALE 0.125f       /* 1/sqrt(64) */

// -------- workspace layout (float offsets) --------
#define WS_QKV_OFF 0                               /* 16*3072          */
#define WS_AO_OFF (WS_QKV_OFF + LQ * 3 * DMODEL)   /* 16*1024          */
#define WS_ACC_OFF (WS_AO_OFF + LQ * DMODEL)       /* H*NSPLIT*L*HDIM  */
#define WS_M_OFF (WS_ACC_OFF + NHEAD * NSPLIT * LQ * HDIM)
#define WS_L_OFF (WS_M_OFF + NHEAD * NSPLIT * LQ)

// ============================================================================
// TDM: issue a 2D tile load (16 rows x 64 f32, row stride in elements) from
// global memory into LDS. D# built per CDNA5 ISA ch.8:
//   group0: [1:0]=count=1, [63:32]=lds_addr, [120:64]=global_addr,
//           [127:126]=type=2
//   group1: [17:16]=data_size=2 (4B), [79:48]=tensor_dim0=64,
//           [111:80]=tensor_dim1=16, [127:112]=tile_dim0=64,
//           [143:128]=tile_dim1=16, [207:160]=tensor_dim0_stride
// Groups 2/3 zero (<=2D tensor). Tracked by TENSORcnt.
// ============================================================================
__device__ __forceinline__ void tdm_load_tile16x64(unsigned lds_byte_addr,
                                                   const float* gsrc,
                                                   unsigned row_stride_elems) {
  const unsigned long long ga = (unsigned long long)(uintptr_t)gsrc;
  u32x4 g0;
  g0[0] = 1u;                                    // count=1, user mode
  g0[1] = lds_byte_addr;                         // LDS destination
  g0[2] = (unsigned)ga;                          // global_addr[31:0]
  g0[3] = (unsigned)((ga >> 32) & 0x01FFFFFFu)   // global_addr[56:32]
          | (2u << 30);                          // type=2 ("image")
  i32x8 g1;
  g1[0] = 0x20000;               // data_size=2 (4B); wg_mask=0 (no cluster)
  g1[1] = (int)(64u << 16);      // tensor_dim0[15:0]=64 in bits[63:48]
  g1[2] = (int)(16u << 16);      // tensor_dim0 hi=0 | tensor_dim1 lo=16
  g1[3] = (int)(64u << 16);      // tensor_dim1 hi=0 | tile_dim0=64
  g1[4] = 16;                    // tile_dim1=16, tile_dim2=0
  g1[5] = (int)row_stride_elems; // tensor_dim0_stride[31:0]
  g1[6] = 0;                     // stride hi | tensor_dim1_stride lo
  g1[7] = 0;
  i32x4 z4 = {0, 0, 0, 0};
#if defined(__clang_major__) && (__clang_major__ >= 23)
  i32x8 z8 = {0, 0, 0, 0, 0, 0, 0, 0};
  __builtin_amdgcn_tensor_load_to_lds(g0, g1, z4, z4, z8, 0);
#else
  __builtin_amdgcn_tensor_load_to_lds(g0, g1, z4, z4, 0);
#endif
}

// Low 32 bits of a flat shared pointer == LDS byte address (LDS aperture).
__device__ __forceinline__ unsigned lds_addr_of(const void* p) {
  return (unsigned)(unsigned long long)(uintptr_t)p;
}

// ============================================================================
// Kernel 1/4: C[16,N] = A[16,K] @ W[K,N] + bias  -- one wave per 16-col tile.
// A-tile:  lane m=lane&15 supplies row m, K pair 2*hi+{0,1}  (float2 load)
// B-tile:  lane n=lane&15 supplies col n, K pair 2*hi+{0,1}  (two strided f32)
// C-tile:  vgpr r -> row r+8*hi, col lane&15
// ============================================================================
__global__ __launch_bounds__(32) void wmma_gemm16(
    const float* __restrict__ A, const float* __restrict__ W,
    const float* __restrict__ bias, float* __restrict__ out, int K, int N) {
  const int lane = threadIdx.x;
  const int m = lane & 15;
  const int hi = lane >> 4;
  const int n0 = blockIdx.x * 16;

  v8f c = {};
#pragma unroll 8
  for (int k0 = 0; k0 < K; k0 += 4) {
    v2f a = *(const v2f*)(A + (size_t)m * K + k0 + 2 * hi);
    v2f b;
    b.x = W[(size_t)(k0 + 2 * hi) * N + n0 + m];
    b.y = W[(size_t)(k0 + 2 * hi + 1) * N + n0 + m];
    c = __builtin_amdgcn_wmma_f32_16x16x4_f32(false, a, false, b, (short)0, c,
                                              false, false);
  }
  const float bv = bias[n0 + m];
#pragma unroll
  for (int r = 0; r < 8; ++r) {
    const int row = r + 8 * hi;
    out[(size_t)row * N + n0 + m] = c[r] + bv;
  }
}

// ============================================================================
// Kernel 2/4: flash-decoding partials. One wave per (head, T-slice).
// TDM streams 16-key K/V tiles into double-buffered LDS (TENSORcnt pipelined);
// Q.K^T via 16x WMMA, online softmax via wave32 shfl_xor(16) row reductions,
// P relayout through LDS, P.V via 16x WMMA. Emits unnormalized acc + (m, l).
// Tiles are 16-aligned and BS==16, so every tile's 16 rows share one physical
// block -> contiguous 2D tile (no gather). New tokens (t >= start_pos) come
// from the qkv workspace (inputs must not be mutated); start_pos % 16 == 0 so
// tiles never straddle the boundary.
// ============================================================================
__global__ __launch_bounds__(32) void attn_partial(
    const float* __restrict__ qkv, const float* __restrict__ k_pool,
    const float* __restrict__ v_pool, const long long* __restrict__ block_ids,
    const int* __restrict__ start_pos_p, float* __restrict__ accp,
    float* __restrict__ mp, float* __restrict__ lp) {
  __shared__ float ldsK[2][16 * 64];
  __shared__ float ldsV[2][16 * 64];
  __shared__ float ldsP[16 * 16];

  const int lane = threadIdx.x;
  const int h = blockIdx.x / NSPLIT;
  const int s = blockIdx.x % NSPLIT;
  const int sp = start_pos_p[0];
  const int m16 = lane & 15;
  const int hi = lane >> 4;

  // Preload Q A-tiles for this head: qa[i] covers dims 4i + 2*hi + {0,1}.
  v2f qa[16];
  {
    const float* qrow = qkv + (size_t)m16 * (3 * DMODEL) + h * HDIM;
#pragma unroll
    for (int i = 0; i < 16; ++i)
      qa[i] = *(const v2f*)(qrow + i * 4 + 2 * hi);
  }

  v8f acc[4] = {};
  float mrow[8], lrow[8];
#pragma unroll
  for (int r = 0; r < 8; ++r) {
    mrow[r] = -__builtin_inff();
    lrow[r] = 0.f;
  }

  const int t_lo = s * CHUNK;
  const int nch = CHUNK / 16; // 64 tiles per slice

  // Issue a K/V TDM pair for tile `ci` into buffer ci&1.
  auto issue_tile = [&](int ci) {
    const int t0 = t_lo + ci * 16;
    const float *kb, *vb;
    unsigned stride;
    if (t0 < sp) {
      const long long blk = block_ids[t0 >> 4]; // uniform -> scalar load
      const size_t base = ((size_t)blk * BLKSZ * NHEAD + (size_t)h) * HDIM;
      kb = k_pool + base;
      vb = v_pool + base;
      stride = NHEAD * HDIM; // 1024 floats between tokens
    } else {
      const float* nrow = qkv + (size_t)(t0 - sp) * (3 * DMODEL) + h * HDIM;
      kb = nrow + DMODEL;     // K slice of qkv
      vb = nrow + 2 * DMODEL; // V slice of qkv
      stride = 3 * DMODEL;    // 3072 floats between tokens
    }
    const int buf = ci & 1;
    tdm_load_tile16x64(lds_addr_of(&ldsK[buf][0]), kb, stride);
    tdm_load_tile16x64(lds_addr_of(&ldsV[buf][0]), vb, stride);
  };

  issue_tile(0);
  for (int ci = 0; ci < nch; ++ci) {
    const int t0 = t_lo + ci * 16;
    if (ci + 1 < nch) {
      issue_tile(ci + 1);                     // stream next pair under compute
      __builtin_amdgcn_s_wait_tensorcnt(2);   // pair `ci` complete (in-order)
    } else {
      __builtin_amdgcn_s_wait_tensorcnt(0);
    }
    asm volatile("" ::: "memory"); // LDS now holds tile `ci`
    const float* bK = &ldsK[ci & 1][0];
    const float* bV = &ldsV[ci & 1][0];

    // ---- scores: S[16,16] = Q @ K^T over d=64 (16 WMMAs of K=4) ----
    v8f sc = {};
#pragma unroll
    for (int kd = 0; kd < 16; ++kd) {
      v2f b = *(const v2f*)(bK + m16 * 64 + kd * 4 + 2 * hi); // K[t0+n][.]
      sc = __builtin_amdgcn_wmma_f32_16x16x4_f32(false, qa[kd], false, b,
                                                 (short)0, sc, false, false);
    }

    // ---- scale + causal mask (only the final tile can be masked) ----
    const bool masked_tile = (t0 + 15 >= sp);
#pragma unroll
    for (int r = 0; r < 8; ++r) {
      const int qrow = r + 8 * hi; // query index of this vgpr's row
      if (masked_tile && (t0 + m16 > sp + qrow))
        sc[r] = -__builtin_inff();
      else
        sc[r] *= ATT_SCALE;
    }

    // ---- online softmax: per-row max/sum across 16 lanes of each half ----
#pragma unroll
    for (int r = 0; r < 8; ++r) {
      float v = sc[r];
#pragma unroll
      for (int o = 8; o >= 1; o >>= 1) v = fmaxf(v, __shfl_xor(v, o, 16));
      const float mn = fmaxf(mrow[r], v);      // finite: >=1 unmasked col/row
      const float corr = __expf(mrow[r] - mn); // 0 on first tile
      const float p = __expf(sc[r] - mn);      // masked -> exp(-inf)=0
      float rs = p;
#pragma unroll
      for (int o = 8; o >= 1; o >>= 1) rs += __shfl_xor(rs, o, 16);
      lrow[r] = lrow[r] * corr + rs;
      mrow[r] = mn;
      acc[0][r] *= corr;
      acc[1][r] *= corr;
      acc[2][r] *= corr;
      acc[3][r] *= corr;
      ldsP[(r + 8 * hi) * 16 + m16] = p; // C layout -> LDS for A relayout
    }
    __syncthreads();

    // ---- P @ V: 4 key sub-tiles (K=4 each) x 4 output column tiles ----
#pragma unroll
    for (int kk = 0; kk < 4; ++kk) {
      v2f a = *(const v2f*)(ldsP + m16 * 16 + kk * 4 + 2 * hi);
#pragma unroll
      for (int n = 0; n < 4; ++n) {
        v2f b;
        b.x = bV[(kk * 4 + 2 * hi) * 64 + n * 16 + m16];
        b.y = bV[(kk * 4 + 2 * hi + 1) * 64 + n * 16 + m16];
        acc[n] = __builtin_amdgcn_wmma_f32_16x16x4_f32(false, a, false, b,
                                                       (short)0, acc[n],
                                                       false, false);
      }
    }
    __syncthreads();
  }

  // ---- emit partials ----
  const int idx = h * NSPLIT + s;
#pragma unroll
  for (int r = 0; r < 8; ++r) {
    const int row = r + 8 * hi;
#pragma unroll
    for (int n = 0; n < 4; ++n)
      accp[(size_t)idx * (LQ * HDIM) + row * HDIM + n * 16 + m16] = acc[n][r];
    if (m16 == 0) { // stats are uniform across each 16-lane half
      mp[idx * LQ + row] = mrow[r];
      lp[idx * LQ + row] = lrow[r];
    }
  }
}

// ============================================================================
// Kernel 3/4: merge NSPLIT partials per head with logsumexp weighting.
// ============================================================================
__global__ __launch_bounds__(256) void attn_combine(
    const float* __restrict__ accp, const float* __restrict__ mp,
    const float* __restrict__ lp, float* __restrict__ attn_out) {
  const int tid = blockIdx.x * 256 + threadIdx.x; // over H*L*HDIM = 16384
  if (tid >= NHEAD * LQ * HDIM) return;
  const int h = tid / (LQ * HDIM);
  const int rem = tid % (LQ * HDIM);
  const int l = rem / HDIM;
  const int dd = rem % HDIM;

  float M = -__builtin_inff();
#pragma unroll
  for (int s = 0; s < NSPLIT; ++s)
    M = fmaxf(M, mp[(h * NSPLIT + s) * LQ + l]);
  float den = 0.f, num = 0.f;
#pragma unroll
  for (int s = 0; s < NSPLIT; ++s) {
    const float w = __expf(mp[(h * NSPLIT + s) * LQ + l] - M);
    den += lp[(h * NSPLIT + s) * LQ + l] * w;
    num += accp[(size_t)(h * NSPLIT + s) * (LQ * HDIM) + l * HDIM + dd] * w;
  }
  attn_out[l * DMODEL + h * HDIM + dd] = num / den;
}

// ============================================================================
extern "C" void kernel_launch(void* const* d_in, const int* in_sizes, int n_in,
                              void* d_out, int out_size, void* d_ws,
                              size_t ws_size, hipStream_t stream) {
  const float* x = (const float*)d_in[0];
  const float* k_pool = (const float*)d_in[1];
  const float* v_pool = (const float*)d_in[2];
  const float* W_attn = (const float*)d_in[3];
  const float* b_attn = (const float*)d_in[4];
  const float* W_proj = (const float*)d_in[5];
  const float* b_proj = (const float*)d_in[6];
  const long long* block_ids = (const long long*)d_in[7];
  const int* start_pos = (const int*)d_in[8]; // low 32 bits of the scalar

  float* ws = (float*)d_ws;
  float* qkv = ws + WS_QKV_OFF;
  float* attn_out = ws + WS_AO_OFF;
  float* accp = ws + WS_ACC_OFF;
  float* mp = ws + WS_M_OFF;
  float* lp = ws + WS_L_OFF;

  // 1) qkv[16,3072] = x @ W_attn + b_attn   (192 single-wave tiles)
  wmma_gemm16<<<(3 * DMODEL) / 16, 32, 0, stream>>>(x, W_attn, b_attn, qkv,
                                                    DMODEL, 3 * DMODEL);
  // 2) flash-decoding partials: 16 heads x 16 T-slices = 256 waves
  attn_partial<<<NHEAD * NSPLIT, 32, 0, stream>>>(qkv, k_pool, v_pool,
                                                  block_ids, start_pos, accp,
                                                  mp, lp);
  // 3) combine splits -> attn_out[16,1024]
  attn_combine<<<(NHEAD * LQ * HDIM) / 256, 256, 0, stream>>>(accp, mp, lp,
                                                              attn_out);
  // 4) out[16,1024] = attn_out @ W_proj + b_proj
  wmma_gemm16<<<DMODEL / 16, 32, 0, stream>>>(attn_out, W_proj, b_proj,
                                              (float*)d_out, DMODEL, DMODEL);
}